// LinearAttention_85822036509398
// MI455X (gfx1250) — compile-verified
//
#include <hip/hip_runtime.h>
#include <math.h>

// ---------------------------------------------------------------------------
// LinearAttention scan for MI455X (gfx1250, wave32).
//
// Shapes: B=2,H=8,NC=32,C=64,O=2,D=128.  Per (b,h,chunk) the reference
// collapses to 4 fp32 GEMMs [64x128]@[128x128] against the LDS-resident
// carry H0, a little elementwise algebra, and a rank-1 carry update at c=63
// followed by row-norm tanh-gelu.  Memory-bound (~170MB @ 23.3TB/s), so we
// keep fp32 and use V_WMMA_F32_16X16X4_F32.
// ---------------------------------------------------------------------------

typedef __attribute__((ext_vector_type(2))) float v2f;
typedef __attribute__((ext_vector_type(8))) float v8f;

#define NB    2
#define NH    8
#define NCHNK 32
#define CLEN  64
#define NORD  2
#define DDIM  128
#define PAD   144          // LDS row stride (floats): 144 mod 64 = 16 -> the two
                           // lane-halves (rows k, k+2) land in disjoint banks.

__global__ __launch_bounds__(1024, 1)
void la_scan_kernel(const float* __restrict__ q,
                    const float* __restrict__ w,
                    const float* __restrict__ u,
                    const float* __restrict__ alpha,
                    const float* __restrict__ hinit,
                    const int*   __restrict__ linact_p,
                    float* __restrict__ out,
                    float* __restrict__ hfinal)
{
    __shared__ float sH[DDIM * PAD];     // carry state H0 (row-major, padded)
    __shared__ float sdww[CLEN];         // dot(w1[c], w0[c])
    __shared__ float sdqw[CLEN];         // dot(q1[c], w0[c])
    __shared__ float sa0[DDIM];          // alpha row c=63, order 0
    __shared__ float sa1[DDIM];          // alpha row c=63, order 1
    __shared__ float suv0[DDIM];         // u_val0 at c=63
    __shared__ float suv1[DDIM];         // u_val1 at c=63
    __shared__ float spart[DDIM * 8];    // per-row sumsq partials

    const int bh   = blockIdx.x;         // 0..15  (b*H + h)
    const int t    = threadIdx.x;        // 0..1023
    const int lane = t & 31;
    const int wave = t >> 5;             // 0..31
    const int half = lane >> 4;          // 0/1
    const int col  = lane & 15;
    const int linact = linact_p[0];

    const size_t seqBase = (size_t)bh * (NCHNK * CLEN * NORD * DDIM);

    // ---- load initial carry state into LDS ----
    {
        const float* hsrc = hinit + (size_t)bh * (DDIM * DDIM);
        for (int i = t; i < DDIM * DDIM; i += 1024) {
            int d = i >> 7, e = i & 127;
            sH[d * PAD + e] = hsrc[i];
        }
    }
    __syncthreads();

    const int eb   = wave & 7;           // e-tile (N) block, 0..7
    const int cb   = wave >> 3;          // c-tile (M) block, 0..3
    const int ecol = eb * 16 + col;      // this lane's N column

    for (int nc = 0; nc < NCHNK; ++nc) {
        const size_t base = seqBase + (size_t)nc * (CLEN * NORD * DDIM);

        // ------- phase 0: per-chunk scalars (dots + alpha row 63) -------
        if (t < CLEN) {
            const float* w0p = w + base + (size_t)t * (NORD * DDIM);
            float acc = 0.f;
            for (int d = 0; d < DDIM; ++d) acc += w0p[DDIM + d] * w0p[d];
            sdww[t] = acc;
        } else if (t < 2 * CLEN) {
            int c = t - CLEN;
            const float* w0p = w + base + (size_t)c * (NORD * DDIM);
            const float* q1p = q + base + (size_t)c * (NORD * DDIM) + DDIM;
            float acc = 0.f;
            for (int d = 0; d < DDIM; ++d) acc += q1p[d] * w0p[d];
            sdqw[c] = acc;
        } else if (t < 2 * CLEN + DDIM) {
            int e = t - 2 * CLEN;
            sa0[e] = alpha[base + (size_t)(CLEN - 1) * (NORD * DDIM) + e];
            sa1[e] = alpha[base + (size_t)(CLEN - 1) * (NORD * DDIM) + DDIM + e];
        }
        // prefetch next chunk's operands (1024 threads x 64B lines = 64KB/tensor)
        if (nc + 1 < NCHNK) {
            const size_t nb = base + (size_t)(CLEN * NORD * DDIM);
            const size_t off = (size_t)t * 16;   // 16 floats = 64B
            __builtin_prefetch(q + nb + off, 0, 3);
            __builtin_prefetch(w + nb + off, 0, 3);
            __builtin_prefetch(u + nb + off, 0, 3);
            __builtin_prefetch(alpha + nb + off, 0, 3);
        }
        __syncthreads();

        // ------- phase 1: WMMA GEMMs  {W0,Q0,W1,Q1} @ H0 -------
        // A fragment (16x4 f32): lane holds row m = cb*16+col, K = half*2 + j.
        // B fragment (4x16 f32): lane holds B[half*2 + j][ecol] from LDS.
        const float* aw = w + base + (size_t)(cb * 16 + col) * (NORD * DDIM);
        const float* aq = q + base + (size_t)(cb * 16 + col) * (NORD * DDIM);
        v8f accW0 = {}; v8f accQ0 = {}; v8f accW1 = {}; v8f accQ1 = {};
        #pragma unroll 4
        for (int k0 = 0; k0 < DDIM; k0 += 4) {
            const int kk = k0 + half * 2;
            v2f bfrag;
            bfrag.x = sH[kk * PAD + ecol];
            bfrag.y = sH[(kk + 1) * PAD + ecol];
            v2f a0f = *(const v2f*)(aw + kk);
            v2f a1f = *(const v2f*)(aq + kk);
            v2f a2f = *(const v2f*)(aw + DDIM + kk);
            v2f a3f = *(const v2f*)(aq + DDIM + kk);
            accW0 = __builtin_amdgcn_wmma_f32_16x16x4_f32(false, a0f, false, bfrag, (short)0, accW0, false, false);
            accQ0 = __builtin_amdgcn_wmma_f32_16x16x4_f32(false, a1f, false, bfrag, (short)0, accQ0, false, false);
            accW1 = __builtin_amdgcn_wmma_f32_16x16x4_f32(false, a2f, false, bfrag, (short)0, accW1, false, false);
            accQ1 = __builtin_amdgcn_wmma_f32_16x16x4_f32(false, a3f, false, bfrag, (short)0, accQ1, false, false);
        }

        // ------- phase 2: elementwise outputs from in-register accumulators ---
        // C/D fragment: element v holds row c = cb*16 + 8*half + v, col ecol.
        #pragma unroll
        for (int v = 0; v < 8; ++v) {
            const int c = cb * 16 + 8 * half + v;
            const size_t i0 = base + (size_t)c * (NORD * DDIM) + ecol;
            const float u0  = u[i0];
            const float q0v = q[i0];
            const float a0e = alpha[i0];
            const float u1  = u[i0 + DDIM];
            const float q1v = q[i0 + DDIM];

            const float uv0 = u0 - accW0[v];                       // u - w0^T H0
            const float o0  = q0v * uv0 + accQ0[v];
            const float k1  = a0e * accW1[v] + sdww[c] * uv0;      // w1^T state1
            const float uv1 = u1 - k1;
            const float o1  = q1v * uv1 + a0e * accQ1[v] + sdqw[c] * uv0;

            out[i0]        = o0;
            out[i0 + DDIM] = o1;
            if (c == CLEN - 1) { suv0[ecol] = uv0; suv1[ecol] = uv1; }
        }
        __syncthreads();

        // ------- phase 3: carry update at c=63 + row-norm gelu -------
        {
            const int d   = t >> 3;        // row of H
            const int ech = t & 7;         // 16-wide column chunk
            const float* wrow = w + base + (size_t)(CLEN - 1) * (NORD * DDIM);
            const float w0r = wrow[d];
            const float w1r = wrow[DDIM + d];
            float sumsq = 0.f;
            #pragma unroll
            for (int j = 0; j < 16; ++j) {
                const int e = ech * 16 + j;
                float h = sH[d * PAD + e];
                h = (h * sa0[e] + w0r * suv0[e]) * sa1[e] + w1r * suv1[e];
                sH[d * PAD + e] = h;
                sumsq += h * h;
            }
            spart[d * 8 + ech] = sumsq;
        }
        __syncthreads();
        {
            const int d   = t >> 3;
            const int ech = t & 7;
            float s = 0.f;
            #pragma unroll
            for (int j = 0; j < 8; ++j) s += spart[d * 8 + j];
            const float norm = sqrtf(s) + 1e-6f;
            if (!linact) {
                const float inv = 2.0f / norm;
                #pragma unroll
                for (int j = 0; j < 16; ++j) {
                    const int e = ech * 16 + j;
                    const float x = sH[d * PAD + e];
                    const float y = x * inv;
                    const float g = 0.5f * y *
                        (1.0f + tanhf(0.7978845608f * (y + 0.044715f * y * y * y)));
                    sH[d * PAD + e] = 0.5f * x * g;
                }
            }
            if (nc == NCHNK - 1) {
                float* hout = hfinal + (size_t)bh * (DDIM * DDIM);
                #pragma unroll
                for (int j = 0; j < 16; ++j) {
                    const int e = ech * 16 + j;
                    hout[d * DDIM + e] = sH[d * PAD + e];
                }
            }
        }
        __syncthreads();
    }
}

extern "C" void kernel_launch(void* const* d_in, const int* in_sizes, int n_in,
                              void* d_out, int out_size, void* d_ws, size_t ws_size,
                              hipStream_t stream) {
    (void)in_sizes; (void)n_in; (void)out_size; (void)d_ws; (void)ws_size;
    const float* q     = (const float*)d_in[0];
    const float* w     = (const float*)d_in[1];
    const float* u     = (const float*)d_in[2];
    const float* alpha = (const float*)d_in[3];
    const float* h0    = (const float*)d_in[4];
    const int*   lin   = (const int*)d_in[5];

    float* out    = (float*)d_out;
    float* hfinal = out + (size_t)NB * NH * NCHNK * CLEN * NORD * DDIM;

    la_scan_kernel<<<dim3(NB * NH), dim3(1024), 0, stream>>>(
        q, w, u, alpha, h0, lin, out, hfinal);
}